// VectorQuantize_6605659701783
// MI455X (gfx1250) — compile-verified
//
#include <hip/hip_runtime.h>

typedef __attribute__((ext_vector_type(2))) float v2f;
typedef __attribute__((ext_vector_type(8))) float v8f;

#define DIM 256
#define NE 8192
#define NROWS 8192
#define NSPLIT 4
#define ROWS_PB 128
#define COLS_PER_SPLIT (NE / NSPLIT)   // 2048
#define COL_CHUNK 64
#define LDS_STRIDE 258                 // 256 + pad(2): even (8B-aligned pairs), breaks bank conflicts
#define EPSV 1e-5f

// ---------------------------------------------------------------------------
// K1: per-code squared norm  eSq[j] = sum_d embed[d][j]^2
// ---------------------------------------------------------------------------
__global__ void vq_esq(const float* __restrict__ embed, float* __restrict__ eSq) {
    int j = blockIdx.x * blockDim.x + threadIdx.x;
    float s = 0.0f;
    for (int d = 0; d < DIM; ++d) {
        float e = embed[(size_t)d * NE + j];   // coalesced over j
        s += e * e;
    }
    eSq[j] = s;
}

// ---------------------------------------------------------------------------
// K2: seed EMA outputs with decay term; zero loss accumulator.
//     cluster_size_new = 0.1*cs (+0.9*bins via atomics later)
//     embed_avg_new    = 0.1*embed_avg (+0.9*embed_sum via atomics later)
// ---------------------------------------------------------------------------
__global__ void vq_init(const float* __restrict__ cs, const float* __restrict__ eavg,
                        float* __restrict__ outCsNew, float* __restrict__ outAvgNew,
                        float* __restrict__ lossRaw) {
    int i = blockIdx.x * blockDim.x + threadIdx.x;
    if (i < DIM * NE) outAvgNew[i] = 0.1f * eavg[i];
    if (i < NE)       outCsNew[i]  = 0.1f * cs[i];
    if (i == 0)       *lossRaw = 0.0f;
}

// ---------------------------------------------------------------------------
// K3: fused f32 WMMA GEMM + row-argmin over a column split.
//     score(i,j) = ||e_j||^2 - 2 * x_i . e_j   (x.x is row-constant)
//     Block: 256 threads (8 waves), owns 128 rows, scans 2048 columns.
// ---------------------------------------------------------------------------
__global__ void __launch_bounds__(256)
vq_dist_argmin(const float* __restrict__ x, const float* __restrict__ embed,
               const float* __restrict__ eSq,
               float* __restrict__ pVal, int* __restrict__ pIdx) {
    extern __shared__ float smem[];
    float* Alds = smem;                           // [ROWS_PB][LDS_STRIDE]
    float* Blds = smem + ROWS_PB * LDS_STRIDE;    // [COL_CHUNK][LDS_STRIDE], transposed [col][k]

    const int t       = threadIdx.x;
    const int rowBase = blockIdx.x * ROWS_PB;
    const int split   = blockIdx.y;
    const int colBase = split * COLS_PER_SPLIT;

    // Stage A: 128 rows x 256 k, fully coalesced
    for (int e = t; e < ROWS_PB * DIM; e += 256) {
        int r = e >> 8;
        int k = e & (DIM - 1);
        Alds[r * LDS_STRIDE + k] = x[(size_t)(rowBase + r) * DIM + k];
    }

    const int lane = t & 31;
    const int wave = t >> 5;
    const int m    = lane & 15;   // M (A-row) / N (B-col) within fragment
    const int h    = lane >> 4;   // half-wave -> K offset of +2

    float minval[8];
    int   minidx[8];
#pragma unroll
    for (int i = 0; i < 8; ++i) { minval[i] = 3.0e38f; minidx[i] = 0; }

    // A fragment base: ISA f32 16x4 layout -> lane reads A[m][k0+2h], A[m][k0+2h+1]
    const float* Arow = &Alds[(wave * 16 + m) * LDS_STRIDE + 2 * h];

    for (int chunk = 0; chunk < COLS_PER_SPLIT / COL_CHUNK; ++chunk) {
        __syncthreads();  // all waves done with previous B tile (and A staged, 1st iter)
        const int c0 = colBase + chunk * COL_CHUNK;
        // Stage B transposed: Blds[col_local][k] = embed[k][c0+col_local]
        for (int e = t; e < COL_CHUNK * DIM; e += 256) {
            int k  = e >> 6;
            int cl = e & (COL_CHUNK - 1);
            Blds[cl * LDS_STRIDE + k] = embed[(size_t)k * NE + c0 + cl];
        }
        __syncthreads();

#pragma unroll
        for (int sub = 0; sub < COL_CHUNK / 16; ++sub) {
            // B fragment: lane reads B[k0+2h][n], B[k0+2h+1][n] == Blds[n][k0+2h..+1]
            const float* Bcol = &Blds[(sub * 16 + m) * LDS_STRIDE + 2 * h];
            v8f acc = {};
#pragma unroll 8
            for (int kk = 0; kk < DIM / 4; ++kk) {
                v2f a = *(const v2f*)(Arow + kk * 4);   // 8B-aligned ds_load_b64
                v2f b = *(const v2f*)(Bcol + kk * 4);
                acc = __builtin_amdgcn_wmma_f32_16x16x4_f32(
                    false, a, false, b, (short)0, acc, false, false);
            }
            const int   col = c0 + sub * 16 + m;
            const float eq  = eSq[col];
#pragma unroll
            for (int i = 0; i < 8; ++i) {
                float v = eq - 2.0f * acc[i];   // acc VGPR i holds rows i (h=0) / i+8 (h=1)
                if (v < minval[i]) { minval[i] = v; minidx[i] = col; }  // '<' keeps first idx
            }
        }
    }

    // Butterfly argmin over the 16 lanes of each half-wave (xor masks stay in-half)
#pragma unroll
    for (int off = 1; off < 16; off <<= 1) {
#pragma unroll
        for (int i = 0; i < 8; ++i) {
            float ov = __shfl_xor(minval[i], off, 32);
            int   oi = __shfl_xor(minidx[i], off, 32);
            if (ov < minval[i] || (ov == minval[i] && oi < minidx[i])) {
                minval[i] = ov; minidx[i] = oi;
            }
        }
    }
    if (m == 0) {  // lane 0 (rows 0-7) and lane 16 (rows 8-15) of each wave
#pragma unroll
        for (int i = 0; i < 8; ++i) {
            int gr = rowBase + wave * 16 + 8 * h + i;
            pVal[(size_t)split * NROWS + gr] = minval[i];
            pIdx[(size_t)split * NROWS + gr] = minidx[i];
        }
    }
}

// ---------------------------------------------------------------------------
// K4: combine the NSPLIT partial argmins per row
// ---------------------------------------------------------------------------
__global__ void vq_reduce_splits(const float* __restrict__ pVal, const int* __restrict__ pIdx,
                                 int* __restrict__ idxOut, float* __restrict__ indOut) {
    int r = blockIdx.x * blockDim.x + threadIdx.x;
    if (r >= NROWS) return;
    float bv = pVal[r];
    int   bi = pIdx[r];
    for (int s = 1; s < NSPLIT; ++s) {
        float v = pVal[(size_t)s * NROWS + r];
        int   i = pIdx[(size_t)s * NROWS + r];
        if (v < bv || (v == bv && i < bi)) { bv = v; bi = i; }
    }
    idxOut[r] = bi;
    indOut[r] = (float)bi;   // argmin index as output element
}

// ---------------------------------------------------------------------------
// K5: gather quantize (== forward value of quantize_st), commit-loss partial
//     reduction, and EMA scatter (bins & embed_sum folded into outputs).
//     One block per row; thread d handles dimension d.
// ---------------------------------------------------------------------------
__global__ void __launch_bounds__(256)
vq_quant_scatter(const float* __restrict__ x, const float* __restrict__ embed,
                 const int* __restrict__ idx,
                 float* __restrict__ outQ, float* __restrict__ outCsNew,
                 float* __restrict__ outAvgNew, float* __restrict__ lossRaw) {
    __shared__ float red[256];
    const int r = blockIdx.x;
    const int d = threadIdx.x;
    const int j = idx[r];
    const float xv = x[(size_t)r * DIM + d];
    const float q  = embed[(size_t)d * NE + j];   // L2-resident (embed = 8MB << 192MB L2)
    outQ[(size_t)r * DIM + d] = q;                // quantize_st forward value == quantize
    const float diff = q - xv;
    red[d] = diff * diff;
    atomicAdd(&outAvgNew[(size_t)d * NE + j], 0.9f * xv);  // (1-decay)*embed_sum
    __syncthreads();
    for (int s = 128; s > 0; s >>= 1) {
        if (d < s) red[d] += red[d + s];
        __syncthreads();
    }
    if (d == 0) {
        atomicAdd(lossRaw, red[0]);
        atomicAdd(&outCsNew[j], 0.9f);            // (1-decay)*bins
    }
}

// ---------------------------------------------------------------------------
// K6: S = sum(cluster_size_new); finalize commit loss (mean * COMMITMENT=1)
// ---------------------------------------------------------------------------
__global__ void vq_sum(const float* __restrict__ csNew, const float* __restrict__ lossRaw,
                       float* __restrict__ Sout, float* __restrict__ outLoss) {
    __shared__ float red[256];
    const int t = threadIdx.x;
    float s = 0.0f;
    for (int i = t; i < NE; i += 256) s += csNew[i];
    red[t] = s;
    __syncthreads();
    for (int st = 128; st > 0; st >>= 1) {
        if (t < st) red[t] += red[t + st];
        __syncthreads();
    }
    if (t == 0) {
        *Sout = red[0];
        *outLoss = *lossRaw * (1.0f / (float)((size_t)NROWS * DIM));
    }
}

// ---------------------------------------------------------------------------
// K7: embed_normalized = embed_avg_new * (S + NE*eps) / ((csn[j]+eps) * S)
// ---------------------------------------------------------------------------
__global__ void vq_normalize(const float* __restrict__ avgNew, const float* __restrict__ csNew,
                             const float* __restrict__ Sp, float* __restrict__ outNorm) {
    int i = blockIdx.x * blockDim.x + threadIdx.x;
    if (i >= DIM * NE) return;
    const float S = *Sp;
    const int j = i & (NE - 1);
    outNorm[i] = avgNew[i] * (S + (float)NE * EPSV) / ((csNew[j] + EPSV) * S);
}

// ---------------------------------------------------------------------------
extern "C" void kernel_launch(void* const* d_in, const int* in_sizes, int n_in,
                              void* d_out, int out_size, void* d_ws, size_t ws_size,
                              hipStream_t stream) {
    const float* x     = (const float*)d_in[0];  // (8,1024,256)
    const float* embed = (const float*)d_in[1];  // (256,8192)
    const float* cs    = (const float*)d_in[2];  // (8192,)
    const float* eavg  = (const float*)d_in[3];  // (256,8192)

    // Outputs concatenated flat in return order
    float* out        = (float*)d_out;
    float* outQ       = out;                          // 2097152
    float* outInd     = outQ + (size_t)NROWS * DIM;   // 8192
    float* outLoss    = outInd + NROWS;               // 1
    float* outNorm    = outLoss + 1;                  // 2097152
    float* outCsNew   = outNorm + (size_t)DIM * NE;   // 8192
    float* outAvgNew  = outCsNew + NE;                // 2097152

    // Workspace layout (~328 KB)
    char*  ws     = (char*)d_ws;
    int*   wsIdx  = (int*)ws;                                   // 8192 ints
    float* wsPVal = (float*)(ws + 32768);                       // NSPLIT*8192 f32
    int*   wsPIdx = (int*)(ws + 32768 + 131072);                // NSPLIT*8192 i32
    float* wsESq  = (float*)(ws + 32768 + 131072 + 131072);     // 8192 f32
    float* wsLoss = (float*)(ws + 32768 + 131072 + 131072 + 32768);
    float* wsS    = wsLoss + 1;

    vq_esq <<<NE / 256, 256, 0, stream>>>(embed, wsESq);
    vq_init<<<(DIM * NE) / 256, 256, 0, stream>>>(cs, eavg, outCsNew, outAvgNew, wsLoss);

    dim3 g3(NROWS / ROWS_PB, NSPLIT);
    size_t ldsBytes = (size_t)(ROWS_PB + COL_CHUNK) * LDS_STRIDE * sizeof(float); // ~198 KB of 320 KB
    vq_dist_argmin<<<g3, 256, ldsBytes, stream>>>(x, embed, wsESq, wsPVal, wsPIdx);

    vq_reduce_splits<<<NROWS / 256, 256, 0, stream>>>(wsPVal, wsPIdx, wsIdx, outInd);
    vq_quant_scatter<<<NROWS, 256, 0, stream>>>(x, embed, wsIdx, outQ, outCsNew, outAvgNew, wsLoss);
    vq_sum<<<1, 256, 0, stream>>>(outCsNew, wsLoss, wsS, outLoss);
    vq_normalize<<<(DIM * NE) / 256, 256, 0, stream>>>(outAvgNew, outCsNew, wsS, outNorm);
}